// DeltaNetLayer_69509750718491
// MI455X (gfx1250) — compile-verified
//
#include <hip/hip_runtime.h>
#include <hip/hip_bf16.h>

// ---------------------------------------------------------------------------
// DeltaNet single-token layer for MI455X (gfx1250, wave32).
// ~342 MB of fp32 weights streamed once -> pure HBM-bandwidth bound
// (floor ~15us at 23.3 TB/s). WMMA f32 16x16x4 is the dot engine:
// one v_wmma consumes a full 4x16 B-tile (256B) loaded with exactly one
// global_load_b64 per lane. Activation vector is staged into LDS by the
// Tensor Data Mover (tensor_load_to_lds + s_wait_tensorcnt).
// ---------------------------------------------------------------------------

#define H_DIM   2048
#define I_DIM   8192
#define QKV_DIM 8192
#define TV_DIM  4096
#define NVH     32
#define NKH     16
#define KD      128
#define VD      128
#define KC_CONV 4
#define PROJ_ROWS (QKV_DIM + TV_DIM + 2 * NVH)   // 12352

typedef float v2f  __attribute__((ext_vector_type(2)));
typedef float v8f  __attribute__((ext_vector_type(8)));
typedef unsigned int u32x4 __attribute__((ext_vector_type(4)));
typedef int   i32x4 __attribute__((ext_vector_type(4)));
typedef int   i32x8 __attribute__((ext_vector_type(8)));

// ---------------- small math helpers ----------------
__device__ __forceinline__ float sigmoidf_(float x) { return 1.0f / (1.0f + __expf(-x)); }
__device__ __forceinline__ float siluf_(float x)    { return x * sigmoidf_(x); }
__device__ __forceinline__ float softplusf_(float x){ return (x > 20.0f) ? x : log1pf(__expf(x)); }

// LDS tree reduction; n = blockDim.x (power of two)
__device__ __forceinline__ float block_sum(float v, float* sh, int n) {
    sh[threadIdx.x] = v;
    __syncthreads();
    for (int s = n >> 1; s > 0; s >>= 1) {
        if ((int)threadIdx.x < s) sh[threadIdx.x] += sh[threadIdx.x + s];
        __syncthreads();
    }
    float r = sh[0];
    __syncthreads();
    return r;
}

// ---------------- TDM: DMA one contiguous row of nfloats into LDS -----------
// D# per CDNA5 ISA ch.8: group0 = {count|flags, lds_addr, global_addr lo,
// global_addr[56:32] | type=2}; group1 = {data_size=4B, tensor_dim0=n,
// tensor_dim1=1, tile_dim0=n, tile_dim1=1, dim0_stride=n}; groups 2/3 zero
// (2D mode). Issued once per block by wave 0; TENSORcnt waited before barrier.
// Toolchain here is the clang-23 lane: 6-arg builtin (extra i32x8 group).
__device__ __forceinline__ void tdm_load_row_to_lds(const float* gsrc,
                                                    float* lds_dst, int nfloats) {
    const unsigned long long ga = (unsigned long long)(uintptr_t)gsrc;
    const unsigned lds_off = (unsigned)(uintptr_t)lds_dst;   // flat LDS addr: low 32b = offset
    u32x4 g0;
    g0.x = 1u;                                      // count=1 valid descriptor
    g0.y = lds_off;                                 // lds_addr (bytes)
    g0.z = (unsigned)(ga & 0xffffffffu);            // global_addr[31:0]
    g0.w = (unsigned)((ga >> 32) & 0x01ffffffu)     // global_addr[56:32]
         | 0x80000000u;                             // type=2 ("image")
    i32x8 g1;
    g1[0] = (int)(2u << 16);                        // data_size = 2 (4 bytes)
    g1[1] = (int)(((unsigned)nfloats & 0xffffu) << 16);   // tensor_dim0[15:0]
    g1[2] = (int)(1u << 16);                        // tensor_dim0 hi=0 | tensor_dim1=1
    g1[3] = (int)((unsigned)nfloats << 16);         // dim1 hi=0 | tile_dim0=n
    g1[4] = 1;                                      // tile_dim1=1 | tile_dim2=0
    g1[5] = nfloats;                                // tensor_dim0_stride[31:0]
    g1[6] = 0;
    g1[7] = 0;
    const i32x4 gz4 = {0, 0, 0, 0};
    const i32x8 gz8 = {0, 0, 0, 0, 0, 0, 0, 0};
    __builtin_amdgcn_tensor_load_to_lds(g0, g1, gz4, gz4, gz8, 0);
}

// ---------------- RMSNorm (n = 2048, one block of 256) ----------------------
__global__ __launch_bounds__(256) void rms_kernel(const float* __restrict__ x,
                                                  const float* __restrict__ w,
                                                  float* __restrict__ out) {
    __shared__ float sh[256];
    float ss = 0.0f;
#pragma unroll
    for (int i = 0; i < H_DIM / 256; ++i) {
        float v = x[threadIdx.x + i * 256];
        ss += v * v;
    }
    ss = block_sum(ss, sh, 256);
    const float inv = rsqrtf(ss / (float)H_DIM + 1e-6f);
#pragma unroll
    for (int i = 0; i < H_DIM / 256; ++i) {
        int idx = threadIdx.x + i * 256;
        out[idx] = x[idx] * inv * (1.0f + w[idx]);
    }
}

// ---------------- WMMA GEMV -------------------------------------------------
// out[r] = dot(W[r,:], vec). One block = 16 output rows; 8 waves split K.
// A-matrix: all 16 rows broadcast the same 4 activations (every D row equals
// the desired dot; we read row 0). Lane layout (per ISA 7.12.2, 32-bit ops):
//   A VGPR0 = {K0 lanes0-15 | K2 lanes16-31}, VGPR1 = {K1 | K3}
//   B VGPR0 = {K0,N=lane    | K2,N=lane-16 }, VGPR1 = {K1 | K3}
// => lane group lo reads (k,k+1), hi reads (k+2,k+3): fold "+2" into the base
// pointer once; inner loop = global_load_b64 + ds_load_b64 + v_wmma.
enum { GEMV_PLAIN = 0, GEMV_RESID = 1, GEMV_GATEUP = 2 };

template <int MODE, int K>
__global__ __launch_bounds__(256) void gemv16_wmma(const float* __restrict__ W,
                                                   const float* __restrict__ W2,
                                                   const float* __restrict__ vec,
                                                   const float* __restrict__ res,
                                                   float* __restrict__ out) {
    constexpr int KCHUNK = K / 8;          // per-wave K slice (mult of 64)
    __shared__ float lds_vec[K];
    __shared__ float red0[8][16];
    __shared__ float red1[8][16];

    // TDM stage of the activation vector into LDS (once per block, wave 0)
    if (threadIdx.x < 32) {
        tdm_load_row_to_lds(vec, lds_vec, K);
        __builtin_amdgcn_s_wait_tensorcnt(0);
    }
    __syncthreads();

    const int lane  = threadIdx.x & 31;
    const int wv    = threadIdx.x >> 5;
    const int rbase = blockIdx.x * 16;
    const int row   = rbase + (lane & 15);
    const int sel   = (lane < 16) ? 0 : 2;   // lo lanes: k,k+1 ; hi: k+2,k+3
    const int k0    = wv * KCHUNK;

    const float* pw  = W + (size_t)row * K + k0 + sel;
    const float* pw2 = (MODE == GEMV_GATEUP) ? (W2 + (size_t)row * K + k0 + sel) : nullptr;
    const float* pv  = lds_vec + k0 + sel;

    v8f acc0 = {};
    v8f acc1 = {};

#pragma unroll 1
    for (int blk = 0; blk < KCHUNK / 64; ++blk) {
        __builtin_prefetch(pw + 512, 0, 1);                 // ~2KB ahead
        if (MODE == GEMV_GATEUP) __builtin_prefetch(pw2 + 512, 0, 1);
#pragma unroll
        for (int u = 0; u < 16; ++u) {
            const float2 av = *reinterpret_cast<const float2*>(pv);
            const float2 wv2 = *reinterpret_cast<const float2*>(pw);
            v2f a; a.x = av.x;  a.y = av.y;
            v2f b; b.x = wv2.x; b.y = wv2.y;
            acc0 = __builtin_amdgcn_wmma_f32_16x16x4_f32(false, a, false, b,
                                                         (short)0, acc0, false, false);
            if (MODE == GEMV_GATEUP) {
                const float2 wv3 = *reinterpret_cast<const float2*>(pw2);
                v2f b2; b2.x = wv3.x; b2.y = wv3.y;
                acc1 = __builtin_amdgcn_wmma_f32_16x16x4_f32(false, a, false, b2,
                                                             (short)0, acc1, false, false);
                pw2 += 4;
            }
            pv += 4;
            pw += 4;
        }
    }
    // D row 0 (M=0) lives in VGPR0 of lanes 0..15 (N = lane)
    if (lane < 16) {
        red0[wv][lane] = acc0[0];
        if (MODE == GEMV_GATEUP) red1[wv][lane] = acc1[0];
    }
    __syncthreads();
    if (threadIdx.x < 16) {
        float s = 0.0f, s2 = 0.0f;
#pragma unroll
        for (int i = 0; i < 8; ++i) {
            s += red0[i][threadIdx.x];
            if (MODE == GEMV_GATEUP) s2 += red1[i][threadIdx.x];
        }
        const int r = rbase + (int)threadIdx.x;
        if (MODE == GEMV_PLAIN)  out[r] = s;
        if (MODE == GEMV_RESID)  out[r] = res[r] + s;
        if (MODE == GEMV_GATEUP) out[r] = siluf_(s) * s2;   // silu(gate)*up
    }
}

// ---------------- conv-state shift + depthwise conv + SiLU ------------------
__global__ __launch_bounds__(256) void conv_kernel(const float* __restrict__ proj,
                                                   const float* __restrict__ conv_state,
                                                   const float* __restrict__ conv_w,
                                                   float* __restrict__ new_conv_out,
                                                   float* __restrict__ conv_out) {
    const int c = blockIdx.x * blockDim.x + threadIdx.x;   // 0..QKV-1
    const float4 cs = *reinterpret_cast<const float4*>(conv_state + c * 4);
    const float4 cw = *reinterpret_cast<const float4*>(conv_w + c * 4);
    const float n3 = proj[c];                              // mixed_qkv[c]
    const float s = cs.y * cw.x + cs.z * cw.y + cs.w * cw.z + n3 * cw.w;
    conv_out[c] = siluf_(s);
    *reinterpret_cast<float4*>(new_conv_out + c * 4) = make_float4(cs.y, cs.z, cs.w, n3);
}

// ---------------- per-head delta rule ---------------------------------------
// One block per value head (32 blocks x 128 threads). Thread t owns column v=t.
// Two passes over the 128x128 head state (64 KB -> L2 resident).
__global__ __launch_bounds__(128) void delta_kernel(const float* __restrict__ conv_out,
                                                    const float* __restrict__ proj,
                                                    const float* __restrict__ ssm_in,
                                                    const float* __restrict__ A_log,
                                                    const float* __restrict__ dt_bias,
                                                    const float* __restrict__ norm_w,
                                                    float* __restrict__ new_ssm,
                                                    float* __restrict__ y_out) {
    __shared__ float qs[KD];
    __shared__ float ks[KD];
    __shared__ float sh[128];
    __shared__ float sc[2];

    const int h  = blockIdx.x;     // value head
    const int t  = threadIdx.x;    // 0..127
    const int kh = h >> 1;         // key/query head (ratio = 2)

    const float qraw = conv_out[kh * KD + t];
    const float kraw = conv_out[NKH * KD + kh * KD + t];
    const float vval = conv_out[2 * NKH * KD + h * VD + t];

    if (t == 0) {
        const float a_raw = proj[QKV_DIM + TV_DIM + h];
        const float b_raw = proj[QKV_DIM + TV_DIM + NVH + h];
        sc[0] = sigmoidf_(b_raw);                                           // beta
        sc[1] = __expf(-__expf(A_log[h]) * softplusf_(a_raw + dt_bias[h])); // decay
    }
    const float qss = block_sum(qraw * qraw, sh, 128);
    const float kss = block_sum(kraw * kraw, sh, 128);
    qs[t] = qraw / fmaxf(sqrtf(qss), 1e-12f) * 0.08838834764831845f;  // l2 * KD^-0.5
    ks[t] = kraw / fmaxf(sqrtf(kss), 1e-12f);
    __syncthreads();

    const float beta  = sc[0];
    const float decay = sc[1];
    const float* S = ssm_in + (size_t)h * KD * VD;
    float*      So = new_ssm + (size_t)h * KD * VD;

    // pass 1: Sk[t] = decay * sum_k S[k][t] * k_norm[k]
    float sk = 0.0f;
#pragma unroll 8
    for (int kk = 0; kk < KD; ++kk) sk += S[kk * VD + t] * ks[kk];
    sk *= decay;
    const float delta = (vval - sk) * beta;

    // pass 2: new state + y = q^T new_state
    float y = 0.0f;
#pragma unroll 8
    for (int kk = 0; kk < KD; ++kk) {
        const float ns = decay * S[kk * VD + t] + ks[kk] * delta;
        So[kk * VD + t] = ns;
        y += ns * qs[kk];
    }

    // gated RMS-norm over VD, then SiLU(z) gate
    const float yss = block_sum(y * y, sh, 128);
    const float yn  = y * rsqrtf(yss / (float)VD + 1e-6f) * norm_w[t];
    const float z   = proj[QKV_DIM + h * VD + t];
    y_out[h * VD + t] = yn * siluf_(z);
}

// ---------------- launch ----------------------------------------------------
extern "C" void kernel_launch(void* const* d_in, const int* in_sizes, int n_in,
                              void* d_out, int out_size, void* d_ws, size_t ws_size,
                              hipStream_t stream) {
    const float* x          = (const float*)d_in[0];
    const float* conv_state = (const float*)d_in[1];
    const float* ssm_state  = (const float*)d_in[2];
    const float* in_proj_w  = (const float*)d_in[3];
    const float* out_proj_w = (const float*)d_in[4];
    const float* conv_w     = (const float*)d_in[5];
    const float* A_log      = (const float*)d_in[6];
    const float* dt_bias    = (const float*)d_in[7];
    const float* norm_w     = (const float*)d_in[8];
    const float* rms1_w     = (const float*)d_in[9];
    const float* rms2_w     = (const float*)d_in[10];
    const float* gate_w     = (const float*)d_in[11];
    const float* up_w       = (const float*)d_in[12];
    const float* down_w     = (const float*)d_in[13];

    float* out        = (float*)d_out;
    float* out_final  = out;                               // 2048
    float* out_conv   = out + H_DIM;                       // 8192*4
    float* out_ssm    = out + H_DIM + QKV_DIM * KC_CONV;   // 32*128*128

    float* ws = (float*)d_ws;
    float* h1   = ws;                   // 2048
    float* proj = h1 + H_DIM;           // 12352
    float* co   = proj + PROJ_ROWS;     // 8192 (conv_out)
    float* yo   = co + QKV_DIM;         // 4096 (gated attn output)
    float* x1   = yo + TV_DIM;          // 2048 (residual 1)
    float* h2   = x1 + H_DIM;           // 2048
    float* gu   = h2 + H_DIM;           // 8192 (silu(gate)*up)

    // 1) h = rms(x, rms1_w)
    rms_kernel<<<1, 256, 0, stream>>>(x, rms1_w, h1);
    // 2) proj = h @ in_proj_w.T          (12352 rows, K=2048)
    gemv16_wmma<GEMV_PLAIN, H_DIM><<<PROJ_ROWS / 16, 256, 0, stream>>>(
        in_proj_w, nullptr, h1, nullptr, proj);
    // 3) conv-state shift + depthwise conv + silu
    conv_kernel<<<QKV_DIM / 256, 256, 0, stream>>>(proj, conv_state, conv_w, out_conv, co);
    // 4) delta rule per value head -> new_ssm, gated y_out
    delta_kernel<<<NVH, 128, 0, stream>>>(co, proj, ssm_state, A_log, dt_bias,
                                          norm_w, out_ssm, yo);
    // 5) x1 = x + y_out @ out_proj_w.T   (2048 rows, K=4096)
    gemv16_wmma<GEMV_RESID, TV_DIM><<<H_DIM / 16, 256, 0, stream>>>(
        out_proj_w, nullptr, yo, x, x1);
    // 6) h2 = rms(x1, rms2_w)
    rms_kernel<<<1, 256, 0, stream>>>(x1, rms2_w, h2);
    // 7) gu = silu(h2 @ gate_w.T) * (h2 @ up_w.T)   (8192 rows, K=2048)
    gemv16_wmma<GEMV_GATEUP, H_DIM><<<I_DIM / 16, 256, 0, stream>>>(
        gate_w, up_w, h2, nullptr, gu);
    // 8) out = x1 + gu @ down_w.T        (2048 rows, K=8192)
    gemv16_wmma<GEMV_RESID, I_DIM><<<H_DIM / 16, 256, 0, stream>>>(
        down_w, nullptr, gu, x1, out_final);
    (void)in_sizes; (void)n_in; (void)out_size; (void)ws_size;
}